// RBFDescriptor_52836687675439
// MI455X (gfx1250) — compile-verified
//
#include <hip/hip_runtime.h>

typedef __attribute__((ext_vector_type(2))) float v2f;
typedef __attribute__((ext_vector_type(8))) float v8f;

#define RBF_NBASIS 16
#define RBF_BETTA  7.1111111f   /* 16^2 / 6^2 */
#define RBF_LOG2E  1.44269504f
/* shifts[k] = 0.5 + (5.5/16)*k = 0.5 + 0.34375*k */

__global__ void rbf_zero_out(float* __restrict__ p, int n) {
  int i = blockIdx.x * blockDim.x + threadIdx.x;
  if (i < n) p[i] = 0.0f;
}

__global__ __launch_bounds__(256) void rbf_descriptor_wmma(
    const float* __restrict__ R, const int* __restrict__ idx0,
    const int* __restrict__ idx1, float* __restrict__ out, int n_edges) {
  const int lane   = threadIdx.x & 31;
  const int wave   = (int)((blockIdx.x * blockDim.x + threadIdx.x) >> 5);
  const int nwaves = (int)((gridDim.x * blockDim.x) >> 5);
  const int nb     = lane & 15;

  // B matrix (4x16 f32), column n = log2(e) * [2*beta*s_n, -beta, -beta*s_n^2, 0]^T
  // (log2e folded in so the WMMA directly produces the v_exp_f32 argument).
  // Layout: VGPR0 = {K0 | K2}, VGPR1 = {K1 | K3} across lane halves.
  const float s = 0.5f + 0.34375f * (float)nb;
  v2f bm;
  bm.x = (lane < 16) ? (RBF_LOG2E * 2.0f * RBF_BETTA * s)
                     : (-RBF_LOG2E * RBF_BETTA * s * s);
  bm.y = (lane < 16) ? (-RBF_LOG2E * RBF_BETTA) : 0.0f;

  const int stride = nwaves * 32;

  for (int base = wave * 32; base < n_edges; base += stride) {
    if (base + 32 <= n_edges) {           // wave-uniform: EXEC all-ones path
      const int e  = base + lane;
      const int i0 = idx0[e];
      const int i1 = idx1[e];

      // prefetch next chunk of the streaming index arrays
      if (base + stride + lane < n_edges) {
        __builtin_prefetch(&idx0[base + stride + lane], 0, 0);
        __builtin_prefetch(&idx1[base + stride + lane], 0, 0);
      }

      // gather positions (L2-resident), compute pair distance (raw v_sqrt_f32)
      const float dx = R[3 * i0 + 0] - R[3 * i1 + 0];
      const float dy = R[3 * i0 + 1] - R[3 * i1 + 1];
      const float dz = R[3 * i0 + 2] - R[3 * i1 + 2];
      const float dr = __builtin_amdgcn_sqrtf(dx * dx + dy * dy + dz * dz);

#pragma unroll
      for (int g = 0; g < 2; ++g) {       // two 16-edge groups per wave-chunk
        // A matrix (16x4 f32): row m = [dr_m, dr_m^2, 1, 0]
        // Layout: lanes 0-15 VGPR0=K0,VGPR1=K1; lanes 16-31 VGPR0=K2,VGPR1=K3
        const float t = __shfl(dr, g * 16 + nb, 32);
        v2f am;
        am.x = (lane < 16) ? t : 1.0f;
        am.y = (lane < 16) ? (t * t) : 0.0f;

        v8f acc = {};
        acc = __builtin_amdgcn_wmma_f32_16x16x4_f32(
            false, am, false, bm, (short)0, acc, false, false);

        // D layout: VGPR r, lanes 0-15 -> edge (g*16+r), lanes 16-31 -> edge (g*16+r+8);
        // lane%16 = basis index. Scatter-add rows into descriptor[idx1[edge]][*].
        const int srcb = g * 16 + ((lane >> 4) << 3);
#pragma unroll
        for (int r = 0; r < 8; ++r) {
          const float val     = __builtin_amdgcn_exp2f(acc[r]);
          const unsigned atom = (unsigned)__shfl(i1, srcb + r, 32);
          const unsigned off  = atom * (unsigned)RBF_NBASIS + (unsigned)nb;
          unsafeAtomicAdd(out + off, val);
        }
      }
    } else {                              // scalar tail (never hit: 6.4M % 256 == 0)
      const int e = base + lane;
      if (e < n_edges) {
        const int i0 = idx0[e], i1 = idx1[e];
        const float dx = R[3 * i0 + 0] - R[3 * i1 + 0];
        const float dy = R[3 * i0 + 1] - R[3 * i1 + 1];
        const float dz = R[3 * i0 + 2] - R[3 * i1 + 2];
        const float dr = __builtin_amdgcn_sqrtf(dx * dx + dy * dy + dz * dz);
#pragma unroll
        for (int k = 0; k < RBF_NBASIS; ++k) {
          const float sk = 0.5f + 0.34375f * (float)k;
          const float d  = sk - dr;
          const unsigned off = (unsigned)i1 * (unsigned)RBF_NBASIS + (unsigned)k;
          unsafeAtomicAdd(out + off,
                          __builtin_amdgcn_exp2f(-RBF_LOG2E * RBF_BETTA * d * d));
        }
      }
    }
  }
}

extern "C" void kernel_launch(void* const* d_in, const int* in_sizes, int n_in,
                              void* d_out, int out_size, void* d_ws, size_t ws_size,
                              hipStream_t stream) {
  const float* R   = (const float*)d_in[0];
  /* d_in[1] = Z: unused by the reference computation */
  const int* idx   = (const int*)d_in[2];
  const int n_edges = in_sizes[2] / 2;
  const int* idx0  = idx;            // idx[0, :]
  const int* idx1  = idx + n_edges;  // idx[1, :]
  float* out = (float*)d_out;

  // Harness poisons d_out; we accumulate with atomics, so zero it first.
  rbf_zero_out<<<(out_size + 255) / 256, 256, 0, stream>>>(out, out_size);

  // 2048 blocks x 256 threads = 16384 waves; ~12 chunks of 32 edges per wave.
  rbf_descriptor_wmma<<<2048, 256, 0, stream>>>(R, idx0, idx1, out, n_edges);
}